// Neuron_50594714747177
// MI455X (gfx1250) — compile-verified
//
#include <hip/hip_runtime.h>

typedef __attribute__((ext_vector_type(2))) float v2f;
typedef __attribute__((ext_vector_type(8))) float v8f;

#define INPUT_DIM 512
#define SIDE_DIM  1024
#define NCTX      8
#define BATCH_N   131072
#define THREADS   256
#define WAVES_PER_BLOCK 8
#define COLS_PER_WAVE   16

__global__ __launch_bounds__(THREADS) void
gln_wmma_kernel(const float* __restrict__ logit_prev,   // [512 x 131072]
                const float* __restrict__ side_info,    // [1024 x 131072]
                const float* __restrict__ vmat,         // [8 x 1024]
                const float* __restrict__ bvec,         // [8]
                const float* __restrict__ weights,      // [256 x 512]
                float* __restrict__ out)                // [131072]
{
    __shared__ float vlds[NCTX * SIDE_DIM];             // 32 KB staged v

    const int tid = threadIdx.x;
    for (int i = tid; i < NCTX * SIDE_DIM; i += THREADS) vlds[i] = vmat[i];
    __syncthreads();

    const int lane = tid & 31;
    const int wave = tid >> 5;
    const int m    = lane & 15;      // A-matrix row this lane carries
    const int kh   = lane >> 4;      // K-half selector (0/1) per ISA VGPR layout
    const int j    = lane & 15;      // column within this wave's 16-col group

    const long jbase = ((long)blockIdx.x * WAVES_PER_BLOCK + wave) * COLS_PER_WAVE;

    // ---- Phase 1: proj(8x16) = v(8x1024) * S(1024, jbase..jbase+15)
    // via V_WMMA_F32_16X16X4_F32, K-loop of 256 steps.
    // A rows m>=8 mirror rows m&7, so C rows 8..15 duplicate rows 0..7 and
    // ALL 32 lanes hold proj[0..7][j] after the loop (no broadcast needed).
    const float* __restrict__ vrow = &vlds[(m & 7) * SIDE_DIM];
    const float* __restrict__ scol = side_info + jbase + j;

    v8f acc = {0.f, 0.f, 0.f, 0.f, 0.f, 0.f, 0.f, 0.f};
#pragma unroll 4
    for (int kk = 0; kk < SIDE_DIM; kk += 4) {
        const int k0 = kk + 2 * kh;
        v2f a, bm;
        a.x  = vrow[k0];                         // ds_read_b64 (adjacent pair)
        a.y  = vrow[k0 + 1];
        bm.x = scol[(long)k0 * BATCH_N];         // coalesced 64B-per-half loads
        bm.y = scol[(long)(k0 + 1) * BATCH_N];
        __builtin_prefetch(scol + (long)(k0 + 16) * BATCH_N, 0, 1);
        acc = __builtin_amdgcn_wmma_f32_16x16x4_f32(
                  /*neg_a=*/false, a, /*neg_b=*/false, bm,
                  /*c_mod=*/(short)0, acc, /*reuse_a=*/false, /*reuse_b=*/false);
    }

    // ---- Context index: bit i set iff proj[i][j] > b[i]
    int ctx = 0;
#pragma unroll
    for (int i = 0; i < NCTX; ++i)
        ctx |= (acc[i] > bvec[i]) ? (1 << i) : 0;

    // ---- Phase 2: out[j] = dot(weights[ctx, :], logit_prev[:, j])
    // Lane halves split the 512-long reduction; L reads are coalesced rows,
    // weights gather stays in L2 (0.5 MB table).
    const float* __restrict__ wrow = weights + (long)ctx * INPUT_DIM;
    const float* __restrict__ lcol = logit_prev + jbase + j;
    float sum = 0.f;
    const int d0 = kh * (INPUT_DIM / 2);
#pragma unroll 8
    for (int d = 0; d < INPUT_DIM / 2; ++d) {
        const int dd = d0 + d;
        sum = fmaf(wrow[dd], lcol[(long)dd * BATCH_N], sum);
    }
    sum += __shfl_xor(sum, 16, 32);              // pairwise cross-half reduce
    if (kh == 0) out[jbase + j] = sum;
}

extern "C" void kernel_launch(void* const* d_in, const int* in_sizes, int n_in,
                              void* d_out, int out_size, void* d_ws, size_t ws_size,
                              hipStream_t stream) {
    const float* logit_prev = (const float*)d_in[0];
    const float* side_info  = (const float*)d_in[1];
    const float* vmat       = (const float*)d_in[2];
    const float* bvec       = (const float*)d_in[3];
    const float* weights    = (const float*)d_in[4];
    float* out = (float*)d_out;

    const int groups = BATCH_N / COLS_PER_WAVE;      // 8192 wave-groups
    const int blocks = groups / WAVES_PER_BLOCK;     // 1024 blocks of 8 waves
    gln_wmma_kernel<<<blocks, THREADS, 0, stream>>>(
        logit_prev, side_info, vmat, bvec, weights, out);
}